// Model_70471823393011
// MI455X (gfx1250) — compile-verified
//
#include <hip/hip_runtime.h>

// ---------------------------------------------------------------------------
// gfx1250 async load-to-LDS path (ASYNCcnt), guarded so compile never fails.
// Builtin signature (probe-confirmed present): parameters are int pointers:
//   void __builtin_amdgcn_global_load_async_to_lds_b32(
//       __attribute__((address_space(1))) int* gsrc,
//       __attribute__((address_space(3))) int* ldst,
//       int imm_offset, int cpol);
// ---------------------------------------------------------------------------
#if defined(__has_builtin)
#  if __has_builtin(__builtin_amdgcn_global_load_async_to_lds_b32) && \
      __has_builtin(__builtin_amdgcn_s_wait_asynccnt)
#    define USE_ASYNC_LDS 1
#  endif
#endif
#ifndef USE_ASYNC_LDS
#  define USE_ASYNC_LDS 0
#endif

typedef __attribute__((address_space(1))) int GInt;
typedef __attribute__((address_space(3))) int LInt;

// ---------------------------------------------------------------------------
// Kernel 1: out_ids[t] — token-parallel gather with per-block LDS copy of
// query_start_loc and a binary search per token.
// ---------------------------------------------------------------------------
__global__ void __launch_bounds__(256)
k_gather_ids(const float* __restrict__ input_ids,
             const int*   __restrict__ idx_mapping,
             const int*   __restrict__ qsl,
             const float* __restrict__ prefill,
             const int*   __restrict__ plen,
             const int*   __restrict__ ncomp,
             float*       __restrict__ out_ids,
             int num_tokens, int nreq, int max_len) {
    extern __shared__ int s_qsl[];  // nreq+1 ints (~2 KB)

#if USE_ASYNC_LDS
    for (int i = (int)threadIdx.x; i <= nreq; i += (int)blockDim.x) {
        __builtin_amdgcn_global_load_async_to_lds_b32(
            (GInt*)(qsl + i), (LInt*)(s_qsl + i), 0, 0);
    }
    __builtin_amdgcn_s_wait_asynccnt(0);
#else
    for (int i = (int)threadIdx.x; i <= nreq; i += (int)blockDim.x) {
        s_qsl[i] = qsl[i];
    }
#endif
    __syncthreads();

    int t = (int)(blockIdx.x * blockDim.x + threadIdx.x);
    if (t >= num_tokens) return;

    // b = largest index with qsl[b] <= t  (searchsorted right - 1)
    int lo = 0, hi = nreq;
    while (lo < hi) {
        int mid = (lo + hi + 1) >> 1;
        if (s_qsl[mid] <= t) lo = mid; else hi = mid - 1;
    }
    int b  = lo;
    int rb = idx_mapping[b];
    int nc = ncomp[rb];
    int pl = plen[rb];

    float v;
    if (nc < pl) {
        int col = nc + (t - s_qsl[b]);
        col = col < 0 ? 0 : (col > max_len - 1 ? max_len - 1 : col);
        v = prefill[(long long)rb * (long long)max_len + (long long)col];
    } else {
        v = input_ids[t];
    }
    out_ids[t] = v;
}

// ---------------------------------------------------------------------------
// Kernel 2: out_next base copy (all MAX_NUM_REQS entries).
// ---------------------------------------------------------------------------
__global__ void __launch_bounds__(256)
k_next_copy(const float* __restrict__ next_prefill,
            float*       __restrict__ out_next,
            int n) {
    int i = (int)(blockIdx.x * blockDim.x + threadIdx.x);
    if (i < n) out_next[i] = next_prefill[i];
}

// ---------------------------------------------------------------------------
// Kernel 3: out_next scatter override for the NUM_REQS mapped requests.
// idx_mapping entries are unique -> no write conflicts.
// ---------------------------------------------------------------------------
__global__ void __launch_bounds__(256)
k_next_scatter(const float* __restrict__ next_prefill,
               const int*   __restrict__ idx_mapping,
               const int*   __restrict__ qsl,
               const float* __restrict__ prefill,
               const int*   __restrict__ plen,
               const int*   __restrict__ ncomp,
               float*       __restrict__ out_next,
               int nreq, int max_len) {
    int i = (int)(blockIdx.x * blockDim.x + threadIdx.x);
    if (i >= nreq) return;

    int r    = idx_mapping[i];
    int nc   = ncomp[r];
    int pl   = plen[r];
    int qlen = qsl[i + 1] - qsl[i];
    int np   = nc + qlen;

    bool write_next = (nc < pl) && (np < pl);
    float v;
    if (write_next) {
        int col = np < 0 ? 0 : (np > max_len - 1 ? max_len - 1 : np);
        v = prefill[(long long)r * (long long)max_len + (long long)col];
    } else {
        v = next_prefill[r];
    }
    out_next[r] = v;
}

// ---------------------------------------------------------------------------
// Launch
// ---------------------------------------------------------------------------
extern "C" void kernel_launch(void* const* d_in, const int* in_sizes, int n_in,
                              void* d_out, int out_size, void* d_ws, size_t ws_size,
                              hipStream_t stream) {
    (void)n_in; (void)out_size; (void)d_ws; (void)ws_size;

    const float* input_ids    = (const float*)d_in[0];  // [NUM_TOKENS]
    const float* next_prefill = (const float*)d_in[1];  // [MAX_NUM_REQS]
    const int*   idx_mapping  = (const int*)  d_in[2];  // [NUM_REQS]
    const int*   qsl          = (const int*)  d_in[3];  // [NUM_REQS+1]
    const float* prefill      = (const float*)d_in[4];  // [MAX_NUM_REQS * MAX_PREFILL_LEN]
    const int*   plen         = (const int*)  d_in[5];  // [MAX_NUM_REQS]
    const int*   ncomp        = (const int*)  d_in[6];  // [MAX_NUM_REQS]

    const int num_tokens = in_sizes[0];
    const int max_reqs   = in_sizes[1];
    const int nreq       = in_sizes[2];
    const int max_len    = in_sizes[4] / in_sizes[5];

    float* out_ids  = (float*)d_out;
    float* out_next = out_ids + num_tokens;

    // out_next: full copy, then ordered scatter override (stream-serialized).
    {
        dim3 blk(256), grd((max_reqs + 255) / 256);
        k_next_copy<<<grd, blk, 0, stream>>>(next_prefill, out_next, max_reqs);
    }
    {
        dim3 blk(256), grd((nreq + 255) / 256);
        k_next_scatter<<<grd, blk, 0, stream>>>(next_prefill, idx_mapping, qsl,
                                                prefill, plen, ncomp,
                                                out_next, nreq, max_len);
    }
    // out_ids: 1 token per lane, fully coalesced reads/writes.
    {
        dim3 blk(256), grd((num_tokens + 255) / 256);
        size_t shmem = (size_t)(nreq + 1) * sizeof(int);
        k_gather_ids<<<grd, blk, shmem, stream>>>(input_ids, idx_mapping, qsl,
                                                  prefill, plen, ncomp,
                                                  out_ids, num_tokens, nreq, max_len);
    }
}